// FaultGAT_86801289052533
// MI455X (gfx1250) — compile-verified
//
#include <hip/hip_runtime.h>
#include <math.h>

typedef float v2f __attribute__((ext_vector_type(2)));
typedef float v8f __attribute__((ext_vector_type(8)));

#define NEG_SLOPE 0.2f

__device__ __forceinline__ void atomicMaxF(float* addr, float val) {
    if (!(val < 0.0f)) {
        atomicMax((int*)addr, __float_as_int(val));
    } else {
        atomicMin((unsigned int*)addr, __float_as_uint(val));
    }
}

__global__ void k_fill(float* p, float v, int n) {
    int i = blockIdx.x * blockDim.x + threadIdx.x;
    if (i < n) p[i] = v;
}

// -------- node linear: hf = x@Wf, hu = x@Wu via V_WMMA_F32_16X16X4_F32 -----
// one wave per 16-row tile; A=16x4 (K=IN=4), B=4x16 col-group of W, D=16x16
__global__ void k_node_linear(const float* __restrict__ x,
                              const float* __restrict__ Wf,
                              const float* __restrict__ Wu,
                              float* __restrict__ hf,
                              float* __restrict__ hu, int N) {
    int tile = blockIdx.x;
    int lane = threadIdx.x;          // 0..31
    int m    = lane & 15;
    int kh   = (lane >> 4) * 2;      // K half: 0 or 2
    int n    = lane & 15;
    int row  = tile * 16 + m;
    bool rv  = row < N;
    v2f a;
    a.x = rv ? x[row * 4 + kh]     : 0.0f;
    a.y = rv ? x[row * 4 + kh + 1] : 0.0f;
    int mrow0 = (lane >> 4) * 8;
    bool full = (tile * 16 + 16) <= N;   // uniform across the wave
    for (int mat = 0; mat < 2; ++mat) {
        const float* W = mat ? Wu : Wf;
        float* hout    = mat ? hu : hf;
#pragma unroll
        for (int g = 0; g < 4; ++g) {
            int cb = g * 16;
            v2f b;
            b.x = W[kh * 64 + cb + n];
            b.y = W[(kh + 1) * 64 + cb + n];
            v8f c = {};
            c = __builtin_amdgcn_wmma_f32_16x16x4_f32(
                    false, a, false, b, (short)0, c, false, false);
            float* dst = hout + (size_t)(tile * 16 + mrow0) * 64 + cb + n;
            if (full) {
#pragma unroll
                for (int v = 0; v < 8; ++v) dst[v * 64] = c[v];
            } else {
#pragma unroll
                for (int v = 0; v < 8; ++v)
                    if (tile * 16 + mrow0 + v < N) dst[v * 64] = c[v];
            }
        }
    }
}

// -------- attention coefficients: wave per (node, mat), lane per channel ----
__global__ void k_alpha(const float* __restrict__ hf, const float* __restrict__ hu,
                        const float* __restrict__ afs, const float* __restrict__ afd,
                        const float* __restrict__ aus, const float* __restrict__ aud,
                        float* asf, float* adf, float* asu, float* adu, int N) {
    int wid  = blockIdx.x * (blockDim.x >> 5) + (threadIdx.x >> 5);
    int lane = threadIdx.x & 31;
    if (wid >= N * 2) return;
    int i = wid >> 1, mat = wid & 1;
    const float* hp  = mat ? hu  : hf;
    const float* aws = mat ? aus : afs;
    const float* awd = mat ? aud : afd;
    float v0 = hp[i * 64 + lane];            // head 0, channel=lane
    float v1 = hp[i * 64 + 32 + lane];       // head 1, channel=lane
    float s0 = v0 * aws[lane],      d0 = v0 * awd[lane];
    float s1 = v1 * aws[32 + lane], d1 = v1 * awd[32 + lane];
#pragma unroll
    for (int o = 16; o > 0; o >>= 1) {
        s0 += __shfl_xor(s0, o, 32);
        d0 += __shfl_xor(d0, o, 32);
        s1 += __shfl_xor(s1, o, 32);
        d1 += __shfl_xor(d1, o, 32);
    }
    if (lane == 0) {
        float* as_o = mat ? asu : asf;
        float* ad_o = mat ? adu : adf;
        as_o[i * 2 + 0] = s0; ad_o[i * 2 + 0] = d0;
        as_o[i * 2 + 1] = s1; ad_o[i * 2 + 1] = d1;
    }
}

__device__ __forceinline__ void edge_sd(const int* ei, int E, int e, int swap,
                                        int& s, int& d) {
    if (e < E) { s = ei[e]; d = ei[E + e]; } else { s = e - E; d = e - E; }
    if (swap) { int t = s; s = d; d = t; }
}

// -------- edge pass 1: segment max of leaky_relu(as[s]+ad[d]) over dst -----
__global__ void k_edge_max(const int* __restrict__ ei, int E, int Etot, int swap,
                           const float* __restrict__ as, const float* __restrict__ ad,
                           float* emax) {
    int t = blockIdx.x * blockDim.x + threadIdx.x;
    if (t >= Etot * 2) return;
    int e = t >> 1, h = t & 1, s, d;
    edge_sd(ei, E, e, swap, s, d);
    float v = as[s * 2 + h] + ad[d * 2 + h];
    v = v > 0.0f ? v : NEG_SLOPE * v;
    atomicMaxF(&emax[d * 2 + h], v);
}

// -------- edge pass 2: denom[d] += exp(e - emax[d]) -------------------------
__global__ void k_edge_expsum(const int* __restrict__ ei, int E, int Etot, int swap,
                              const float* __restrict__ as, const float* __restrict__ ad,
                              const float* __restrict__ emax, float* den) {
    int t = blockIdx.x * blockDim.x + threadIdx.x;
    if (t >= Etot * 2) return;
    int e = t >> 1, h = t & 1, s, d;
    edge_sd(ei, E, e, swap, s, d);
    float v = as[s * 2 + h] + ad[d * 2 + h];
    v = v > 0.0f ? v : NEG_SLOPE * v;
    atomicAdd(&den[d * 2 + h], expf(v - emax[d * 2 + h]));
}

// -------- edge pass 3: out[d] += h[s] * alpha; lane-per-channel, coalesced --
__global__ void k_edge_agg(const int* __restrict__ ei, int E, int Etot, int swap,
                           const float* __restrict__ as, const float* __restrict__ ad,
                           const float* __restrict__ emax, const float* __restrict__ den,
                           const float* __restrict__ hsrc, float* out) {
    int t = blockIdx.x * blockDim.x + threadIdx.x;
    if (t >= Etot * 64) return;
    int e = t >> 6;            // one edge per 64 threads (2 waves)
    int c = t & 63;            // channel 0..63; head = c>>5
    int h = c >> 5;
    int s, d;
    edge_sd(ei, E, e, swap, s, d);
    int idx = d * 2 + h;
    float v = as[s * 2 + h] + ad[idx];
    v = v > 0.0f ? v : NEG_SLOPE * v;
    float alpha = expf(v - emax[idx]) / (den[idx] + 1e-16f);
    atomicAdd(&out[d * 64 + c], hsrc[s * 64 + c] * alpha);
}

__global__ void k_bias_relu(float* p, const float* __restrict__ bias, int total) {
    int t = blockIdx.x * blockDim.x + threadIdx.x;
    if (t >= total) return;
    float v = p[t] + bias[t & 63];
    p[t] = v > 0.0f ? v : 0.0f;
}

// -------- FC: h2 = relu([outf|outu] @ Wfc + bfc), K=128 as 32 K=4 WMMAs ----
__global__ void k_fc(const float* __restrict__ outf, const float* __restrict__ outu,
                     const float* __restrict__ Wfc, const float* __restrict__ bfc,
                     float* __restrict__ h2, int N) {
    int tile = blockIdx.x;
    int lane = threadIdx.x;
    int m    = lane & 15;
    int n    = lane & 15;
    int row  = tile * 16 + m;
    bool rv  = row < N;
    v8f acc0 = {}, acc1 = {}, acc2 = {}, acc3 = {};
    for (int kk = 0; kk < 32; ++kk) {
        int k0 = kk * 4 + (lane >> 4) * 2;   // even; pair stays on one side of 64
        const float* src = (k0 < 64) ? (outf + (size_t)row * 64 + k0)
                                     : (outu + (size_t)row * 64 + (k0 - 64));
        v2f a;
        a.x = rv ? src[0] : 0.0f;
        a.y = rv ? src[1] : 0.0f;
        v2f b;
        b.x = Wfc[k0 * 64 + 0 + n];  b.y = Wfc[(k0 + 1) * 64 + 0 + n];
        acc0 = __builtin_amdgcn_wmma_f32_16x16x4_f32(false, a, false, b, (short)0, acc0, false, false);
        b.x = Wfc[k0 * 64 + 16 + n]; b.y = Wfc[(k0 + 1) * 64 + 16 + n];
        acc1 = __builtin_amdgcn_wmma_f32_16x16x4_f32(false, a, false, b, (short)0, acc1, false, false);
        b.x = Wfc[k0 * 64 + 32 + n]; b.y = Wfc[(k0 + 1) * 64 + 32 + n];
        acc2 = __builtin_amdgcn_wmma_f32_16x16x4_f32(false, a, false, b, (short)0, acc2, false, false);
        b.x = Wfc[k0 * 64 + 48 + n]; b.y = Wfc[(k0 + 1) * 64 + 48 + n];
        acc3 = __builtin_amdgcn_wmma_f32_16x16x4_f32(false, a, false, b, (short)0, acc3, false, false);
    }
    int mrow0 = (lane >> 4) * 8;
    bool full = (tile * 16 + 16) <= N;   // uniform across the wave
    float* dst = h2 + (size_t)(tile * 16 + mrow0) * 64 + n;
    if (full) {
#pragma unroll
        for (int v = 0; v < 8; ++v) {
            float t0 = acc0[v] + bfc[0 + n];  dst[v * 64 + 0]  = t0 > 0 ? t0 : 0;
            float t1 = acc1[v] + bfc[16 + n]; dst[v * 64 + 16] = t1 > 0 ? t1 : 0;
            float t2 = acc2[v] + bfc[32 + n]; dst[v * 64 + 32] = t2 > 0 ? t2 : 0;
            float t3 = acc3[v] + bfc[48 + n]; dst[v * 64 + 48] = t3 > 0 ? t3 : 0;
        }
    } else {
#pragma unroll
        for (int v = 0; v < 8; ++v) {
            if (tile * 16 + mrow0 + v < N) {
                float t0 = acc0[v] + bfc[0 + n];  dst[v * 64 + 0]  = t0 > 0 ? t0 : 0;
                float t1 = acc1[v] + bfc[16 + n]; dst[v * 64 + 16] = t1 > 0 ? t1 : 0;
                float t2 = acc2[v] + bfc[32 + n]; dst[v * 64 + 32] = t2 > 0 ? t2 : 0;
                float t3 = acc3[v] + bfc[48 + n]; dst[v * 64 + 48] = t3 > 0 ? t3 : 0;
            }
        }
    }
}

// -------- final conv: h3 = h2@Wo (wave per node, coalesced + shuffle) -------
__global__ void k_final_lin(const float* __restrict__ h2, const float* __restrict__ Wo,
                            const float* __restrict__ aos, const float* __restrict__ aod,
                            float* h3, float* as3, float* ad3, int N) {
    int wid  = blockIdx.x * (blockDim.x >> 5) + (threadIdx.x >> 5);
    int lane = threadIdx.x & 31;
    if (wid >= N) return;
    float p = h2[(size_t)wid * 64 + lane] * Wo[lane]
            + h2[(size_t)wid * 64 + 32 + lane] * Wo[32 + lane];
#pragma unroll
    for (int o = 16; o > 0; o >>= 1) p += __shfl_xor(p, o, 32);
    if (lane == 0) {
        h3[wid]  = p;
        as3[wid] = p * aos[0];
        ad3[wid] = p * aod[0];
    }
}

__global__ void k3_max(const int* __restrict__ ei, int E, int Etot,
                       const float* __restrict__ as3, const float* __restrict__ ad3,
                       float* emax3) {
    int e = blockIdx.x * blockDim.x + threadIdx.x;
    if (e >= Etot) return;
    int s, d; edge_sd(ei, E, e, 0, s, d);
    float v = as3[s] + ad3[d];
    v = v > 0.0f ? v : NEG_SLOPE * v;
    atomicMaxF(&emax3[d], v);
}

__global__ void k3_expsum(const int* __restrict__ ei, int E, int Etot,
                          const float* __restrict__ as3, const float* __restrict__ ad3,
                          const float* __restrict__ emax3, float* den3) {
    int e = blockIdx.x * blockDim.x + threadIdx.x;
    if (e >= Etot) return;
    int s, d; edge_sd(ei, E, e, 0, s, d);
    float v = as3[s] + ad3[d];
    v = v > 0.0f ? v : NEG_SLOPE * v;
    atomicAdd(&den3[d], expf(v - emax3[d]));
}

__global__ void k3_agg(const int* __restrict__ ei, int E, int Etot,
                       const float* __restrict__ as3, const float* __restrict__ ad3,
                       const float* __restrict__ emax3, const float* __restrict__ den3,
                       const float* __restrict__ h3, float* out3) {
    int e = blockIdx.x * blockDim.x + threadIdx.x;
    if (e >= Etot) return;
    int s, d; edge_sd(ei, E, e, 0, s, d);
    float v = as3[s] + ad3[d];
    v = v > 0.0f ? v : NEG_SLOPE * v;
    float alpha = expf(v - emax3[d]) / (den3[d] + 1e-16f);
    atomicAdd(&out3[d], h3[s] * alpha);
}

__global__ void k_sigmoid(const float* __restrict__ out3, const float* __restrict__ bo,
                          float* out, int N) {
    int i = blockIdx.x * blockDim.x + threadIdx.x;
    if (i >= N) return;
    float v = out3[i] + bo[0];
    out[i] = 1.0f / (1.0f + expf(-v));
}

extern "C" void kernel_launch(void* const* d_in, const int* in_sizes, int n_in,
                              void* d_out, int out_size, void* d_ws, size_t ws_size,
                              hipStream_t stream) {
    const float* x   = (const float*)d_in[0];
    const int*   ei  = (const int*)  d_in[1];
    const float* Wf  = (const float*)d_in[2];
    const float* afs = (const float*)d_in[3];
    const float* afd = (const float*)d_in[4];
    const float* bf  = (const float*)d_in[5];
    const float* Wu  = (const float*)d_in[6];
    const float* aus = (const float*)d_in[7];
    const float* aud = (const float*)d_in[8];
    const float* bu  = (const float*)d_in[9];
    const float* Wfc = (const float*)d_in[10];
    const float* bfc = (const float*)d_in[11];
    const float* Wo  = (const float*)d_in[12];
    const float* aos = (const float*)d_in[13];
    const float* aod = (const float*)d_in[14];
    const float* bo  = (const float*)d_in[15];
    float* out = (float*)d_out;

    const int N    = in_sizes[0] / 4;
    const int E    = in_sizes[1] / 2;
    const int Etot = E + N;

    // workspace carving (floats)
    float* ws = (float*)d_ws;
    size_t off = 0;
    auto carve = [&](size_t n) { float* p = ws + off; off += n; return p; };
    float* hf    = carve((size_t)N * 64);
    float* hu    = carve((size_t)N * 64);
    float* outf  = carve((size_t)N * 64);
    float* outu  = carve((size_t)N * 64);
    float* asf   = carve((size_t)N * 2);
    float* adf   = carve((size_t)N * 2);
    float* asu   = carve((size_t)N * 2);
    float* adu   = carve((size_t)N * 2);
    float* emaxf = carve((size_t)N * 2);
    float* denf  = carve((size_t)N * 2);
    float* emaxu = carve((size_t)N * 2);
    float* denu  = carve((size_t)N * 2);
    float* h3    = carve(N);
    float* as3   = carve(N);
    float* ad3   = carve(N);
    float* emax3 = carve(N);
    float* den3  = carve(N);
    float* out3  = carve(N);
    float* h2    = hf;   // hf is dead by the time the FC layer runs

    const float NEG_INF = -INFINITY;
    const int TB = 256;
    auto blk = [&](long long n) { return (int)((n + TB - 1) / TB); };

    // init accumulators (must happen every call; harness poisons buffers)
    k_fill<<<blk(N * 64), TB, 0, stream>>>(outf, 0.0f, N * 64);
    k_fill<<<blk(N * 64), TB, 0, stream>>>(outu, 0.0f, N * 64);
    k_fill<<<blk(N * 2), TB, 0, stream>>>(emaxf, NEG_INF, N * 2);
    k_fill<<<blk(N * 2), TB, 0, stream>>>(denf, 0.0f, N * 2);
    k_fill<<<blk(N * 2), TB, 0, stream>>>(emaxu, NEG_INF, N * 2);
    k_fill<<<blk(N * 2), TB, 0, stream>>>(denu, 0.0f, N * 2);
    k_fill<<<blk(N), TB, 0, stream>>>(emax3, NEG_INF, N);
    k_fill<<<blk(N), TB, 0, stream>>>(den3, 0.0f, N);
    k_fill<<<blk(N), TB, 0, stream>>>(out3, 0.0f, N);

    const int tiles = (N + 15) / 16;

    // node linear (WMMA f32 16x16x4) + attention coefficients
    k_node_linear<<<tiles, 32, 0, stream>>>(x, Wf, Wu, hf, hu, N);
    k_alpha<<<blk((long long)N * 2 * 32), TB, 0, stream>>>(hf, hu, afs, afd, aus, aud,
                                                           asf, adf, asu, adu, N);

    // forward conv (swap=0): src -> dst
    k_edge_max   <<<blk((long long)Etot * 2), TB, 0, stream>>>(ei, E, Etot, 0, asf, adf, emaxf);
    k_edge_expsum<<<blk((long long)Etot * 2), TB, 0, stream>>>(ei, E, Etot, 0, asf, adf, emaxf, denf);
    k_edge_agg   <<<blk((long long)Etot * 64), TB, 0, stream>>>(ei, E, Etot, 0, asf, adf, emaxf, denf, hf, outf);

    // upstream conv (swap=1): dst -> src
    k_edge_max   <<<blk((long long)Etot * 2), TB, 0, stream>>>(ei, E, Etot, 1, asu, adu, emaxu);
    k_edge_expsum<<<blk((long long)Etot * 2), TB, 0, stream>>>(ei, E, Etot, 1, asu, adu, emaxu, denu);
    k_edge_agg   <<<blk((long long)Etot * 64), TB, 0, stream>>>(ei, E, Etot, 1, asu, adu, emaxu, denu, hu, outu);

    // bias + relu on both conv outputs
    k_bias_relu<<<blk(N * 64), TB, 0, stream>>>(outf, bf, N * 64);
    k_bias_relu<<<blk(N * 64), TB, 0, stream>>>(outu, bu, N * 64);

    // FC layer (WMMA f32, K=128 chained) -> h2 (aliases hf)
    k_fc<<<tiles, 32, 0, stream>>>(outf, outu, Wfc, bfc, h2, N);

    // final 1-head GAT conv
    k_final_lin<<<blk((long long)N * 32), TB, 0, stream>>>(h2, Wo, aos, aod, h3, as3, ad3, N);
    k3_max   <<<blk(Etot), TB, 0, stream>>>(ei, E, Etot, as3, ad3, emax3);
    k3_expsum<<<blk(Etot), TB, 0, stream>>>(ei, E, Etot, as3, ad3, emax3, den3);
    k3_agg   <<<blk(Etot), TB, 0, stream>>>(ei, E, Etot, as3, ad3, emax3, den3, h3, out3);
    k_sigmoid<<<blk(N), TB, 0, stream>>>(out3, bo, out, N);
}